// NashCascadeNetwork_90314572300545
// MI455X (gfx1250) — compile-verified
//
#include <hip/hip_runtime.h>
#include <hip/hip_bf16.h>
#include <stdint.h>

// Nash cascade network on MI455X (gfx1250, wave32).
// Serial spigot recurrence per bucket -> critical-path bound, not BW bound
// (176 MB @ 23.3 TB/s ~ 7.6us << serial chain). Strategy:
//  * 4-op dependent chain per step: v_fma -> v_max -> v_sqrt -> v_fma,
//    with e_s = H0 - sp_h and c_s = th*a*sqrt(2G) computed off-chain.
//  * CDNA5 async global->LDS loads (ASYNCcnt) double-buffer the 12 B/step
//    operand stream so fetches never sit on the serial chain.
//  * Column sums via wave32 shuffle tree + ds_add_f32, off-chain.

#define NB 1024      // buckets per mid layer == spigots per bucket
#define NM 14        // mid layers
#define NL 16        // total layers (head + mids + outlet)
#define TILE 16      // spigot steps staged per tile
#define BK 128       // buckets per workgroup (4 waves of 32)
#define TWO_G 19.6f
#define SQ2G 4.42718872424f   // sqrt(19.6)

// ---------- CDNA5 async global->LDS helpers (inline asm; ASYNCcnt) ----------
__device__ __forceinline__ uint32_t lds_off(const void* p) {
  // Low 32 bits of a generic pointer into LDS == DS byte address.
  return (uint32_t)(uintptr_t)p;
}

__device__ __forceinline__ void async_ld_b128(uint32_t lds_addr, const void* gptr) {
  asm volatile("global_load_async_to_lds_b128 %0, %1, off"
               :
               : "v"(lds_addr), "v"(gptr)
               : "memory");
}

__device__ __forceinline__ void wait_async0() {
  asm volatile("s_wait_asynccnt 0" ::: "memory");
}

// ---------- layer 0: single head bucket, 1024 spigots, serial ----------
__global__ __launch_bounds__(256) void layer0_kernel(
    const float* __restrict__ H0, const float* __restrict__ S0,
    const float* __restrict__ th0, const float* __restrict__ precip,
    float* __restrict__ out0, float* __restrict__ inflow0) {
  __shared__ float se[NB];   // e_s = H + (s==0 ? p_layer : 0) - sp_h
  __shared__ float sc[NB];   // c_s = th*a*sqrt(2G)
  const int tid = threadIdx.x;
  const float H = H0[0];
  const float pl = precip[0] * (1.0f / (float)NL);
  for (int s = tid; s < NB; s += 256) {
    float e = H - S0[2 * s];
    if (s == 0) e += pl;                   // inflow only affects spigot 0
    se[s] = e;
    sc[s] = th0[s] * S0[2 * s + 1] * SQ2G;
  }
  __syncthreads();
  if (tid == 0) {
    const float pb = pl * (1.0f / (float)NB);
    float cum = 0.0f;
    for (int s = 0; s < NB; ++s) {
      float h = fmaxf(0.0f, fmaf(-0.5f, cum, se[s]));
      float sq = sqrtf(h);
      float c = sc[s];
      inflow0[s] = fmaf(c, sq, pb);        // q_s + p_bucket
      cum = fmaf(c, sq, cum);
    }
    out0[0] = H - cum + pl;                // H0_new
  }
}

// ---------- per-layer inflow init: inflow_next[s] = p_bucket ----------
__global__ void init_inflow_kernel(float* __restrict__ buf,
                                   const float* __restrict__ precip) {
  const int s = blockIdx.x * blockDim.x + threadIdx.x;
  if (s < NB) buf[s] = precip[0] * (1.0f / ((float)NL * (float)NB));
}

// ---------- mid layer: 1024 buckets x 1024 serial spigots ----------
__global__ __launch_bounds__(BK) void mid_layer_kernel(
    const float* __restrict__ H_l,        // (NB,)
    const float* __restrict__ S_l,        // (NB, NB, 2)
    const float* __restrict__ Th_l,       // (NB, NB)
    const float* __restrict__ inflow_in,  // (NB,)
    float* __restrict__ inflow_next,      // (NB,) pre-init to p_bucket
    float* __restrict__ H_out) {          // (NB,) new heads
  // Double-buffered operand tiles + per-block column-sum accumulator.
  __shared__ __align__(16) float sS[2][BK * TILE * 2];  // 32 KB
  __shared__ __align__(16) float sT[2][BK * TILE];      // 16 KB
  __shared__ float colsum[NB];                          // 4 KB

  const int tid = threadIdx.x;
  const int b = blockIdx.x * BK + tid;        // this thread's bucket
  const int lane = tid & 31;

  for (int s = tid; s < NB; s += BK) colsum[s] = 0.0f;

  const char* gS = (const char*)(S_l + (size_t)b * (NB * 2));
  const char* gT = (const char*)(Th_l + (size_t)b * NB);

  // Stage tile t of this thread's bucket row into buffer bb (async).
  auto stage = [&](int t, int bb) {
    uint32_t lS = lds_off(&sS[bb][tid * TILE * 2]);
    const char* pS = gS + (size_t)t * (TILE * 2 * 4);   // 128 B contiguous
#pragma unroll
    for (int k = 0; k < 8; ++k) async_ld_b128(lS + k * 16, pS + k * 16);
    uint32_t lT = lds_off(&sT[bb][tid * TILE]);
    const char* pT = gT + (size_t)t * (TILE * 4);       // 64 B contiguous
#pragma unroll
    for (int k = 0; k < 4; ++k) async_ld_b128(lT + k * 16, pT + k * 16);
  };

  const float H0b = H_l[b];
  const float inb = inflow_in[b];
  float cum = 0.0f;

  stage(0, 0);
  wait_async0();
  __syncthreads();

  for (int t = 0; t < NB / TILE; ++t) {
    const int cur = t & 1;
    if (t + 1 < NB / TILE) stage(t + 1, cur ^ 1);   // prefetch next tile

    const float* tS = &sS[cur][tid * TILE * 2];
    const float* tT = &sT[cur][tid * TILE];
#pragma unroll
    for (int k = 0; k < TILE; ++k) {
      // Off-chain operand prep (no dependence on cum):
      float e = H0b - tS[2 * k];                 // e_s = H0 - sp_h
      if (k == 0 && t == 0) e += inb;            // inflow hits spigot 0 only
      float c = tT[k] * tS[2 * k + 1] * SQ2G;    // c_s = th*a*sqrt(2G)
      // Critical chain (4 dependent ops): fma -> max -> sqrt -> fma
      float h = fmaxf(0.0f, fmaf(-0.5f, cum, e));
      float sq = sqrtf(h);
      cum = fmaf(c, sq, cum);
      // Column sum over buckets (off the critical path): wave32 tree + LDS add
      float qs = c * sq;
      qs += __shfl_xor(qs, 16);
      qs += __shfl_xor(qs, 8);
      qs += __shfl_xor(qs, 4);
      qs += __shfl_xor(qs, 2);
      qs += __shfl_xor(qs, 1);
      if (lane == 0) atomicAdd(&colsum[t * TILE + k], qs);
    }

    wait_async0();       // next tile resident in LDS
    __syncthreads();     // all waves done with buffers + colsum for this tile
  }

  H_out[b] = H0b - cum + inb;

  // Fold this block's partial column sums into the global inflow vector.
  for (int s = tid; s < NB; s += BK) atomicAdd(&inflow_next[s], colsum[s]);
}

// ---------- last layer: 1024 buckets, one outlet spigot each ----------
__global__ __launch_bounds__(256) void last_layer_kernel(
    const float* __restrict__ H_last, const float* __restrict__ S_last,
    const float* __restrict__ th_last, const float* __restrict__ inflow,
    float* __restrict__ Hout, float* __restrict__ qout) {
  const int b = blockIdx.x * blockDim.x + threadIdx.x;
  if (b < NB) {
    float H = H_last[b];
    float inb = inflow[b];
    float sph = S_last[2 * b];
    float spa = S_last[2 * b + 1];
    float h = fmaxf(0.0f, H + inb - sph);
    float q = th_last[b] * spa * SQ2G * sqrtf(h);
    Hout[b] = H - q + inb;
    qout[b] = q;
  }
}

extern "C" void kernel_launch(void* const* d_in, const int* in_sizes, int n_in,
                              void* d_out, int out_size, void* d_ws, size_t ws_size,
                              hipStream_t stream) {
  const float* H0      = (const float*)d_in[0];
  const float* H_mid   = (const float*)d_in[1];
  const float* H_last  = (const float*)d_in[2];
  const float* S0      = (const float*)d_in[3];
  const float* S_mid   = (const float*)d_in[4];
  const float* S_last  = (const float*)d_in[5];
  const float* th0     = (const float*)d_in[6];
  const float* th_mid  = (const float*)d_in[7];
  const float* th_last = (const float*)d_in[8];
  const float* precip  = (const float*)d_in[9];

  float* out = (float*)d_out;
  float* buf0 = (float*)d_ws;          // inflow ping
  float* buf1 = buf0 + NB;             // inflow pong

  // Layer 0: head bucket -> buf0 holds inflow into mid layer 0.
  layer0_kernel<<<1, 256, 0, stream>>>(H0, S0, th0, precip, out, buf0);

  // 14 mid layers, serialized by stream order.
  for (int l = 0; l < NM; ++l) {
    float* in_b  = (l & 1) ? buf1 : buf0;
    float* out_b = (l & 1) ? buf0 : buf1;
    init_inflow_kernel<<<4, 256, 0, stream>>>(out_b, precip);
    mid_layer_kernel<<<NB / BK, BK, 0, stream>>>(
        H_mid + (size_t)l * NB,
        S_mid + (size_t)l * NB * NB * 2,
        th_mid + (size_t)l * NB * NB,
        in_b, out_b,
        out + 1 + (size_t)l * NB);
  }

  // Outlet layer reads buf0 (layer 13 is odd -> wrote buf0).
  last_layer_kernel<<<4, 256, 0, stream>>>(
      H_last, S_last, th_last, buf0,
      out + 1 + (size_t)NM * NB,
      out + 1 + (size_t)NM * NB + NB);
}